// LogitSelector_91113436217852
// MI455X (gfx1250) — compile-verified
//
#include <hip/hip_runtime.h>
#include <stdint.h>

#define RANK 100
#define NUM_ITEMS 6714
#define BATCH 8192
#define THREADS 256

typedef __attribute__((ext_vector_type(4))) unsigned int v4u;
typedef __attribute__((ext_vector_type(8))) int v8i;
typedef __attribute__((ext_vector_type(4))) int v4i;

#if defined(__AMDGCN__) && __has_builtin(__builtin_amdgcn_tensor_load_to_lds)
#define USE_TDM 1
#else
#define USE_TDM 0
#endif

// Order-preserving bijection float-bits -> u32 (larger ord == larger float)
__device__ __forceinline__ uint32_t ord_of(uint32_t u) {
  return u ^ ((u & 0x80000000u) ? 0xFFFFFFFFu : 0x80000000u);
}
__device__ __forceinline__ uint32_t ord_inv(uint32_t o) {
  return (o & 0x80000000u) ? (o ^ 0x80000000u) : ~o;
}

__global__ __launch_bounds__(THREADS)
void LogitSelector_topk_kernel(const float* __restrict__ out_mat,
                               const int* __restrict__ labels,
                               float* __restrict__ d_new_out,
                               float* __restrict__ d_new_lab) {
  __shared__ uint32_t vals[NUM_ITEMS];   // row, as ord-transformed keys
  __shared__ uint32_t hist[256];
  __shared__ uint64_t skey[128];
  __shared__ uint32_t cand_val[RANK];
  __shared__ uint32_t cand_idx[RANK];
  __shared__ uint32_t eq_idx[128];
  __shared__ uint32_t sh_prefix;
  __shared__ int      sh_remaining;
  __shared__ uint32_t sh_cnt_gt, sh_cnt_eq;
  __shared__ int      sh_label_pos;

  const int row = blockIdx.x;
  const int tid = threadIdx.x;
  const float* src = out_mat + (size_t)row * NUM_ITEMS;

  // ---- Stage 1: row -> LDS ----
#if USE_TDM
  if (tid < 32) {  // one wave issues the TDM DMA
    uint64_t ga = (uint64_t)(uintptr_t)src;
    uint32_t la = (uint32_t)(uintptr_t)(&vals[0]);  // low 32 bits = LDS byte offset
    v4u g0;
    g0.x = 1u;                                          // count=1 (valid), user mode
    g0.y = la;                                          // lds_addr
    g0.z = (uint32_t)(ga & 0xFFFFFFFFu);                // global_addr[31:0]
    g0.w = (uint32_t)((ga >> 32) & 0x1FFFFFFu) | 0x80000000u; // addr[56:32] | type=2
    v8i g1;
    g1[0] = (int)(2u << 16);                            // data_size=2 (4B), wg_mask=0
    g1[1] = (int)((uint32_t)NUM_ITEMS << 16);           // tensor_dim0 lo16 (hi bits in g1[2])
    g1[2] = (int)(((uint32_t)NUM_ITEMS >> 16) | (1u << 16)); // dim0 hi16 | tensor_dim1 lo16 = 1
    g1[3] = (int)((uint32_t)NUM_ITEMS << 16);           // tensor_dim1 hi=0 | tile_dim0=6714
    g1[4] = 1;                                          // tile_dim1=1, tile_dim2=0
    g1[5] = NUM_ITEMS;                                  // tensor_dim0_stride lo32
    g1[6] = (int)((uint32_t)NUM_ITEMS << 16);           // stride0 hi=0 | stride1 lo16
    g1[7] = 0;                                          // stride1 hi bits
    v4i g2 = {0, 0, 0, 0};
    v4i g3 = {0, 0, 0, 0};
#if __has_include(<hip/amd_detail/amd_gfx1250_TDM.h>)
    v8i gz = {0, 0, 0, 0, 0, 0, 0, 0};
    __builtin_amdgcn_tensor_load_to_lds(g0, g1, g2, g3, gz, 0);
#else
    __builtin_amdgcn_tensor_load_to_lds(g0, g1, g2, g3, 0);
#endif
    __builtin_amdgcn_s_wait_tensorcnt(0);
  }
  __syncthreads();
  for (int i = tid; i < NUM_ITEMS; i += THREADS) vals[i] = ord_of(vals[i]);
#else
  for (int i = tid; i < NUM_ITEMS; i += THREADS)
    vals[i] = ord_of(__float_as_uint(src[i]));
#endif
  if (tid == 0) {
    sh_prefix = 0; sh_remaining = RANK;
    sh_cnt_gt = 0; sh_cnt_eq = 0; sh_label_pos = -1;
  }
  __syncthreads();

  // ---- Stage 2: exact radix-select of the RANK-th largest key ----
  for (int shift = 24; shift >= 0; shift -= 8) {
    hist[tid] = 0;
    __syncthreads();
    const uint32_t prefix = sh_prefix;
    for (int i = tid; i < NUM_ITEMS; i += THREADS) {
      uint32_t v = vals[i];
      bool match = (shift == 24) || (((v ^ prefix) >> (shift + 8)) == 0);
      if (match) atomicAdd(&hist[(v >> shift) & 0xFFu], 1u);
    }
    __syncthreads();
    if (tid == 0) {
      int rem = sh_remaining;
      int b = 255;
      for (; b > 0; --b) {
        int c = (int)hist[b];
        if (rem <= c) break;
        rem -= c;
      }
      sh_remaining = rem;
      sh_prefix = prefix | ((uint32_t)b << shift);
    }
    __syncthreads();
  }
  const uint32_t vstar = sh_prefix;     // exact key of 100th-largest element
  const int r_eq = sh_remaining;        // how many ==vstar belong to top-100

  // ---- Stage 3: collect candidates ----
  for (int i = tid; i < NUM_ITEMS; i += THREADS) {
    uint32_t v = vals[i];
    if (v > vstar) {
      uint32_t p = atomicAdd(&sh_cnt_gt, 1u);
      cand_val[p] = v; cand_idx[p] = (uint32_t)i;
    } else if (v == vstar) {
      uint32_t p = atomicAdd(&sh_cnt_eq, 1u);
      if (p < 128u) eq_idx[p] = (uint32_t)i;
    }
  }
  __syncthreads();

  if (tid == 0) {
    int ne = (int)sh_cnt_eq; if (ne > 128) ne = 128;
    for (int a = 1; a < ne; ++a) {      // sort tie indices ascending
      uint32_t x = eq_idx[a]; int b = a - 1;
      while (b >= 0 && eq_idx[b] > x) { eq_idx[b + 1] = eq_idx[b]; --b; }
      eq_idx[b + 1] = x;
    }
    uint32_t base = sh_cnt_gt;          // == RANK - r_eq
    for (int t = 0; t < r_eq; ++t) { cand_val[base + t] = vstar; cand_idx[base + t] = eq_idx[t]; }
  }
  __syncthreads();

  // ---- Stage 4: bitonic sort 128 keys descending ((ord<<32)|~idx) ----
  if (tid < 128)
    skey[tid] = (tid < RANK)
        ? (((uint64_t)cand_val[tid] << 32) | (uint32_t)~cand_idx[tid])
        : 0ull;
  __syncthreads();
  for (int k = 2; k <= 128; k <<= 1) {
    for (int j = k >> 1; j > 0; j >>= 1) {
      if (tid < 128) {
        int ixj = tid ^ j;
        if (ixj > tid) {
          uint64_t a = skey[tid], b = skey[ixj];
          bool desc = ((tid & k) == 0);
          if (desc ? (a < b) : (a > b)) { skey[tid] = b; skey[ixj] = a; }
        }
      }
      __syncthreads();
    }
  }

  // ---- Stage 5: reversed gather + label handling ----
  const int lab = labels[row];
  if (tid < RANK) {
    uint64_t e = skey[(RANK - 1) - tid];          // ascending order
    uint32_t idx = ~(uint32_t)(e & 0xFFFFFFFFu);
    uint32_t vo  = (uint32_t)(e >> 32);
    d_new_out[(size_t)row * RANK + tid] = __uint_as_float(ord_inv(vo));
    if ((int)idx == lab) sh_label_pos = tid;      // indices unique: one writer
  }
  __syncthreads();
  if (tid == 0) {
    int pos = sh_label_pos;
    if (pos < 0) {                                 // label missed top-100
      d_new_out[(size_t)row * RANK] = __uint_as_float(ord_inv(vals[lab]));
      pos = 0;
    }
    d_new_lab[row] = (float)pos;
  }
}

extern "C" void kernel_launch(void* const* d_in, const int* in_sizes, int n_in,
                              void* d_out, int out_size, void* d_ws, size_t ws_size,
                              hipStream_t stream) {
  (void)in_sizes; (void)n_in; (void)d_ws; (void)ws_size; (void)out_size;
  const float* outm  = (const float*)d_in[0];
  const int*   labs  = (const int*)d_in[1];
  float* new_out = (float*)d_out;                       // BATCH*RANK values
  float* new_lab = (float*)d_out + (size_t)BATCH * RANK; // BATCH labels (as f32)
  LogitSelector_topk_kernel<<<BATCH, THREADS, 0, stream>>>(outm, labs, new_out, new_lab);
}